// HiPPO_LegS_11304353923244
// MI455X (gfx1250) — compile-verified
//
#include <hip/hip_runtime.h>

// ---------------------------------------------------------------------------
// HiPPO-LegS scan for MI455X (gfx1250, wave32)
//
//   x_t = A_t x_{t-1} + B_t * u_t          (per step: 256x256 GEMM on state)
//
// Memory floor: 256MB A-stream + 256MB output ~= 22us @ 23.3 TB/s; compute
// need is only ~1.6 TFLOP/s -> fp32 WMMA 16x16x4 is the correct precision
// (keeps the 1024-step recurrence fp32-exact vs reference, plenty fast).
//
// Decomposition: 16 workgroups, each owns a 16-batch slice and runs the whole
// time loop privately (no inter-WG sync). All WGs stream the same A_t, so the
// 192MB L2 serves the broadcast; HBM reads A once. State lives transposed in
// LDS (pitch 17 floats -> conflict-free columns), double buffered.
// ---------------------------------------------------------------------------

typedef __attribute__((ext_vector_type(2))) float v2f;
typedef __attribute__((ext_vector_type(8))) float v8f;

#define NSTATE   256
#define LSEQ     1024
#define BATCH    256
#define BTILE    16     // batch rows per workgroup
#define LDSPITCH 17     // padded row stride (floats) for x^T in LDS

__global__ __launch_bounds__(256, 1)
void hippo_legs_scan_f32wmma(const float* __restrict__ inp,   // (L, B)
                             const float* __restrict__ A,     // (L, N, N)
                             const float* __restrict__ Bst,   // (L, N)
                             float*       __restrict__ out)   // (L, B, N)
{
  __shared__ float xT[2][NSTATE * LDSPITCH];   // 2 x 17,408 B

  const int tid = threadIdx.x;
  const int l   = tid & 31;       // lane in wave32
  const int w   = tid >> 5;       // wave 0..7
  const int bl  = l & 15;         // low-lane index (batch col / A-row offset)
  const int hi  = l >> 4;         // half-wave select (K pair for A/B operands)
  const int hi2 = hi * 2;
  const int b0  = blockIdx.x * BTILE;

  const int n0   = w * 32;        // this wave's 32 output-state rows
  const int rowA0 = n0 + bl;      // tile 0 A-operand row (lanes 0-15 == 16-31)
  const int rowA1 = n0 + 16 + bl; // tile 1

  // x_{-1} = 0 in both buffers
  for (int i = tid; i < NSTATE * LDSPITCH; i += 256) {
    xT[0][i] = 0.0f;
    xT[1][i] = 0.0f;
  }
  __syncthreads();

  int cur = 0;
  for (int t = 0; t < LSEQ; ++t) {
    const int nxt = cur ^ 1;
    const float* At = A   + (size_t)t * (NSTATE * NSTATE);
    const float* Bt = Bst + (size_t)t * NSTATE;

    // u_t for this lane's batch column (lanes 16-31 broadcast-duplicate 0-15)
    const float u_b = inp[(size_t)t * BATCH + b0 + bl];

    // Accumulators seeded with B_t * u_t  (C/D layout: lane = batch col,
    // VGPR r = state row n0 + hi*8 + r)
    v8f acc0, acc1;
    {
      const float* p0 = Bt + n0 + hi * 8;
      const float* p1 = Bt + n0 + 16 + hi * 8;
#pragma unroll
      for (int r = 0; r < 8; ++r) {
        acc0[r] = p0[r] * u_b;
        acc1[r] = p1[r] * u_b;
      }
    }

    // A-operand 16x4 fp32 layout: lanes 0-15 hold K=kk+0/1, lanes 16-31
    // hold K=kk+2/3, M striped over lanes&15 -> per-lane b64 from global.
    const float* arow0 = At + (size_t)rowA0 * NSTATE + hi2;
    const float* arow1 = At + (size_t)rowA1 * NSTATE + hi2;
    // B-operand 4x16 fp32: row K striped over lanes within a VGPR; element
    // B[k][b] = xT[k*17 + b] in LDS.
    const float* xrow = &xT[cur][hi2 * LDSPITCH + bl];

#pragma unroll 8
    for (int kk = 0; kk < NSTATE; kk += 4) {
      v2f a0 = *(const v2f*)(arow0 + kk);
      v2f a1 = *(const v2f*)(arow1 + kk);
      v2f bb;
      bb[0] = xrow[(size_t)kk * LDSPITCH];
      bb[1] = xrow[(size_t)(kk + 1) * LDSPITCH];
      acc0 = __builtin_amdgcn_wmma_f32_16x16x4_f32(
          false, a0, false, bb, (short)0, acc0, false, false);
      acc1 = __builtin_amdgcn_wmma_f32_16x16x4_f32(
          false, a1, false, bb, (short)0, acc1, false, false);
    }

    // Publish x_t^T to the other LDS buffer for the next step's B-operands.
#pragma unroll
    for (int r = 0; r < 8; ++r) {
      xT[nxt][(n0 + hi * 8 + r) * LDSPITCH + bl]      = acc0[r];
      xT[nxt][(n0 + 16 + hi * 8 + r) * LDSPITCH + bl] = acc1[r];
    }

    // Prefetch next step's A slice: one lane per state row, 8 x 128B per row
    // -> global_prefetch_b8 covers the wave's full 32KB next-step slice.
    if (t + 1 < LSEQ) {
      const float* An = At + NSTATE * NSTATE + (size_t)(n0 + l) * NSTATE;
#pragma unroll
      for (int j = 0; j < 8; ++j)
        __builtin_prefetch(An + j * 32, 0, 1);
    }

    __syncthreads();   // x_t^T complete; also fences reuse of buffer 'cur'

    // Coalesced emit of x_t: thread -> (batch col, 16-float n chunk); 16
    // consecutive threads write one contiguous 1KB row of out[t, b, :].
    {
      const int bcol = tid >> 4;          // 0..15
      const int nb   = (tid & 15) * 16;   // n chunk base
      float4* dst = (float4*)(out + (((size_t)t * BATCH) + b0 + bcol) * NSTATE + nb);
#pragma unroll
      for (int j = 0; j < 4; ++j) {
        float4 v;
        v.x = xT[nxt][(nb + j * 4 + 0) * LDSPITCH + bcol];
        v.y = xT[nxt][(nb + j * 4 + 1) * LDSPITCH + bcol];
        v.z = xT[nxt][(nb + j * 4 + 2) * LDSPITCH + bcol];
        v.w = xT[nxt][(nb + j * 4 + 3) * LDSPITCH + bcol];
        dst[j] = v;
      }
    }

    cur = nxt;
  }
}

extern "C" void kernel_launch(void* const* d_in, const int* in_sizes, int n_in,
                              void* d_out, int out_size, void* d_ws, size_t ws_size,
                              hipStream_t stream) {
  (void)in_sizes; (void)n_in; (void)out_size; (void)d_ws; (void)ws_size;
  const float* inp = (const float*)d_in[0];   // (L, B)    = 1024 x 256
  const float* A   = (const float*)d_in[1];   // (L, N, N) = 1024 x 256 x 256
  const float* Bst = (const float*)d_in[2];   // (L, N)    = 1024 x 256
  float* out = (float*)d_out;                 // (L, B, N)

  hippo_legs_scan_f32wmma<<<dim3(BATCH / BTILE), dim3(256), 0, stream>>>(
      inp, A, Bst, out);
}